// Attention3D_23948737643122
// MI455X (gfx1250) — compile-verified
//
#include <hip/hip_runtime.h>
#include <math.h>

// ---------------------------------------------------------------------------
// Attention3D (channel attention) for MI455X / gfx1250, fp32 end-to-end.
// Matrix stages use V_WMMA_F32_16X16X4_F32 (fp32 WMMA, wave32).
// ---------------------------------------------------------------------------

typedef __attribute__((ext_vector_type(2))) float v2f;
typedef __attribute__((ext_vector_type(8))) float v8f;

#define WMMA_F32(a, b, c) \
  __builtin_amdgcn_wmma_f32_16x16x4_f32(false, (a), false, (b), (short)0, (c), false, false)

static constexpr int  B_     = 8;
static constexpr int  C_     = 64;     // DIM
static constexpr int  C3_    = 192;    // 3*DIM
static constexpr int  HEADS_ = 8;
static constexpr int  CH_    = 8;      // DIM / HEADS
static constexpr int  S_     = 32768;  // N*H*W = 32*32*32
static constexpr int  NT     = 4;      // 16x16 n-tiles per wave
static constexpr int  WPB    = 8;      // waves per block (256 threads)
static constexpr int  CHUNKS = 32;     // S-chunks for q.k^T partials
static constexpr int  SCHUNK = S_ / CHUNKS;

// ---------------------------------------------------------------------------
// Per-batch GEMM:  Y[b] (M x S) = W[b] (M x 64, row-major) * X[b] (64 x S)
// One wave owns a 16x64 strip: 4 accumulators, K-loop of 16 -> 64 WMMAs/wave.
// fp32 WMMA layouts (ISA 7.12.2):
//   A 16x4 : lane L holds row m=L&15; VGPRs {K=khi, K=khi+1}, khi=2*(L>>4)
//   B 4x16 : lane L holds col n=L&15; VGPRs {K=khi, K=khi+1}
//   C/D    : lane L col n=L&15; VGPR r holds row r + 8*(L>>4)
// ---------------------------------------------------------------------------
__global__ __launch_bounds__(256)
void pw_gemm_wmma(const float* __restrict__ W, long w_bstride,
                  const float* __restrict__ X, long x_bstride,
                  float* __restrict__ Y, long y_bstride) {
  const int lane = threadIdx.x & 31;
  const int wave = threadIdx.x >> 5;
  const int b    = blockIdx.z;
  const float* Wb = W + (long)b * w_bstride;
  const float* Xb = X + (long)b * x_bstride;
  float*       Yb = Y + (long)b * y_bstride;

  const int mbase  = blockIdx.y * 16;
  const int ncol   = lane & 15;          // m-row for A, n-col for B/C/D
  const int khi    = (lane >> 4) << 1;   // 0 or 2
  const int hi8    = (lane >> 4) << 3;   // 0 or 8
  const int ntile0 = (blockIdx.x * WPB + wave) * NT;

  v8f acc[NT] = {};

  for (int k0 = 0; k0 < C_; k0 += 4) {
    v2f a;
    const float* wrow = Wb + (long)(mbase + ncol) * C_ + k0 + khi;
    a.x = wrow[0];
    a.y = wrow[1];
#pragma unroll
    for (int t = 0; t < NT; ++t) {
      const int n = (ntile0 + t) * 16 + ncol;
      v2f bv;
      bv.x = Xb[(long)(k0 + khi)     * S_ + n];
      bv.y = Xb[(long)(k0 + khi + 1) * S_ + n];
      acc[t] = WMMA_F32(a, bv, acc[t]);
    }
  }
#pragma unroll
  for (int t = 0; t < NT; ++t) {
    const int n = (ntile0 + t) * 16 + ncol;
#pragma unroll
    for (int r = 0; r < 8; ++r)
      Yb[(long)(mbase + hi8 + r) * S_ + n] = acc[t][r];
  }
}

// ---------------------------------------------------------------------------
// Depthwise 3x3x3 conv, padding SAME, on (B, 192, 32, 32, 32).
// ---------------------------------------------------------------------------
__global__ __launch_bounds__(256)
void dwconv333(const float* __restrict__ in, const float* __restrict__ wdw,
               float* __restrict__ out) {
  const long idx = (long)blockIdx.x * blockDim.x + threadIdx.x;
  const int x = (int)(idx & 31), y = (int)((idx >> 5) & 31), z = (int)((idx >> 10) & 31);
  const long bc = idx >> 15;                 // b*192 + c
  const int  c  = (int)(bc % C3_);
  const float* w = wdw + c * 27;
  const float* p = in + (bc << 15);
  float s = 0.f;
#pragma unroll
  for (int dz = -1; dz <= 1; ++dz) {
    const int zz = z + dz;
    if (zz < 0 || zz > 31) continue;
#pragma unroll
    for (int dy = -1; dy <= 1; ++dy) {
      const int yy = y + dy;
      if (yy < 0 || yy > 31) continue;
#pragma unroll
      for (int dx = -1; dx <= 1; ++dx) {
        const int xx = x + dx;
        if (xx < 0 || xx > 31) continue;
        s += w[(dz + 1) * 9 + (dy + 1) * 3 + (dx + 1)] * p[(zz << 10) + (yy << 5) + xx];
      }
    }
  }
  out[idx] = s;
}

// ---------------------------------------------------------------------------
// 1 / max(||row||_2, eps) for q (which=0) and k (which=1) rows of length S.
// blockIdx.x = which*512 + b*64 + channel
// ---------------------------------------------------------------------------
__global__ __launch_bounds__(256)
void inv_l2norm(const float* __restrict__ qkv, float* __restrict__ invn) {
  const int id = blockIdx.x;
  const int cf = id & 63, b = (id >> 6) & 7, which = id >> 9;
  const float* p = qkv + ((long)b * C3_ + which * C_ + cf) * S_;
  float s = 0.f;
  for (int i = threadIdx.x; i < S_; i += 256) { const float v = p[i]; s += v * v; }
  __shared__ float red[256];
  red[threadIdx.x] = s;
  __syncthreads();
  for (int off = 128; off > 0; off >>= 1) {
    if (threadIdx.x < off) red[threadIdx.x] += red[threadIdx.x + off];
    __syncthreads();
  }
  if (threadIdx.x == 0) invn[id] = 1.0f / fmaxf(sqrtf(red[0]), 1e-12f);
}

// ---------------------------------------------------------------------------
// Q_bh (8 x S) * K_bh^T via fp32 WMMA (rows/cols 8..15 zero-padded by lane
// masks as multiplies -> uniform control flow, EXEC all-1s for WMMA).
// One wave per (b, h, S-chunk); writes an 8x8 partial Gram matrix.
// ---------------------------------------------------------------------------
__global__ __launch_bounds__(32)
void qk_dots_wmma(const float* __restrict__ qkv, float* __restrict__ partial) {
  const int id    = blockIdx.x;
  const int chunk = id & (CHUNKS - 1);
  const int bh    = id / CHUNKS;              // b*8 + h
  const int h     = bh & 7, b = bh >> 3;
  const int lane  = threadIdx.x;
  const int mn    = lane & 15;
  const int khi   = (lane >> 4) << 1;
  const float msk = (mn < CH_) ? 1.0f : 0.0f;
  const int  row  = (mn < CH_) ? mn : 0;      // always-valid address, masked data
  const float* qrow = qkv + ((long)b * C3_ +        h * CH_ + row) * S_;
  const float* krow = qkv + ((long)b * C3_ + C_ +   h * CH_ + row) * S_;
  const int s0base = chunk * SCHUNK;

  v8f acc = {};
  for (int s = 0; s < SCHUNK; s += 4) {
    const int s0 = s0base + s + khi;
    v2f a, bv;
    a.x  = qrow[s0]     * msk;
    a.y  = qrow[s0 + 1] * msk;
    bv.x = krow[s0]     * msk;
    bv.y = krow[s0 + 1] * msk;
    acc = WMMA_F32(a, bv, acc);
  }
  // D rows 0..7 live in lanes 0..15 (hi half holds rows 8..15 = garbage/zero)
  if (lane < CH_) {
    float* dst = partial + (long)id * 64;
#pragma unroll
    for (int r = 0; r < CH_; ++r) dst[r * 8 + lane] = acc[r];  // [c=r][d=lane]
  }
}

// ---------------------------------------------------------------------------
// Reduce chunk partials, scale by invnorms * temperature[b] (temperature
// broadcasts over batch: b == heads in this module), stable softmax over d.
// One thread per (b, h, c) row; 512 rows total.
// ---------------------------------------------------------------------------
__global__ __launch_bounds__(256)
void softmax_attn(const float* __restrict__ partial, const float* __restrict__ invn,
                  const float* __restrict__ temp, float* __restrict__ attn) {
  const int row = blockIdx.x * blockDim.x + threadIdx.x;
  if (row >= B_ * HEADS_ * CH_) return;
  const int c = row & 7, h = (row >> 3) & 7, b = row >> 6;
  const int bh = b * 8 + h;
  const float iq = invn[b * 64 + h * 8 + c];
  const float t  = temp[b];
  float logit[8];
#pragma unroll
  for (int d = 0; d < 8; ++d) {
    float s = 0.f;
    for (int ch = 0; ch < CHUNKS; ++ch)
      s += partial[((long)bh * CHUNKS + ch) * 64 + c * 8 + d];
    logit[d] = s * iq * invn[512 + b * 64 + h * 8 + d] * t;
  }
  float mx = logit[0];
#pragma unroll
  for (int d = 1; d < 8; ++d) mx = fmaxf(mx, logit[d]);
  float den = 0.f;
#pragma unroll
  for (int d = 0; d < 8; ++d) { logit[d] = __expf(logit[d] - mx); den += logit[d]; }
  const float inv = 1.0f / den;
  float* dst = attn + (long)row * 8;
#pragma unroll
  for (int d = 0; d < 8; ++d) dst[d] = logit[d] * inv;
}

// ---------------------------------------------------------------------------
// W_eff[b] = w_proj * blockdiag(attn[b]) : fuses attn@v with the output 1x1x1
// projection into a single 64x64-per-batch GEMM weight.
// ---------------------------------------------------------------------------
__global__ __launch_bounds__(256)
void build_weff(const float* __restrict__ wproj, const float* __restrict__ attn,
                float* __restrict__ weff) {
  const int b = blockIdx.x;
  for (int e = threadIdx.x; e < C_ * C_; e += 256) {
    const int o = e >> 6, df = e & 63, h = df >> 3, d = df & 7;
    float s = 0.f;
#pragma unroll
    for (int c = 0; c < 8; ++c)
      s += wproj[o * 64 + h * 8 + c] * attn[(((long)b * 8 + h) * 8 + c) * 8 + d];
    weff[(long)b * 4096 + e] = s;
  }
}

// ---------------------------------------------------------------------------
extern "C" void kernel_launch(void* const* d_in, const int* in_sizes, int n_in,
                              void* d_out, int out_size, void* d_ws, size_t ws_size,
                              hipStream_t stream) {
  const float* x      = (const float*)d_in[0];  // (8, 64, 32, 32, 32)
  const float* w_qkv  = (const float*)d_in[1];  // (192, 64)
  const float* w_dw   = (const float*)d_in[2];  // (192, 27)
  const float* w_proj = (const float*)d_in[3];  // (64, 64)
  const float* temp   = (const float*)d_in[4];  // (8,)
  float* out = (float*)d_out;                   // (8, 64, 32768)
  float* ws  = (float*)d_ws;

  float* qkv_pre = ws;                               // 8*192*32768
  float* qkv     = qkv_pre + (long)B_ * C3_ * S_;    // 8*192*32768
  float* invn    = qkv     + (long)B_ * C3_ * S_;    // 2*8*64
  float* partial = invn    + 1024;                   // 64*32*64
  float* attn    = partial + (long)64 * CHUNKS * 64; // 8*8*8*8
  float* weff    = attn    + 4096;                   // 8*64*64

  // 1) qkv_pre[b] = w_qkv (192x64) @ x[b] (64xS)        [WMMA f32]
  dim3 g1(S_ / (16 * NT * WPB), C3_ / 16, B_);
  pw_gemm_wmma<<<g1, 256, 0, stream>>>(w_qkv, 0L, x, (long)C_ * S_, qkv_pre, (long)C3_ * S_);

  // 2) depthwise 3x3x3 conv
  const long total = (long)B_ * C3_ * S_;
  dwconv333<<<dim3((unsigned)(total / 256)), 256, 0, stream>>>(qkv_pre, w_dw, qkv);

  // 3) inverse L2 norms of q and k rows
  inv_l2norm<<<dim3(2 * B_ * C_), 256, 0, stream>>>(qkv, invn);

  // 4) q.k^T Gram matrices                               [WMMA f32]
  qk_dots_wmma<<<dim3(B_ * HEADS_ * CHUNKS), 32, 0, stream>>>(qkv, partial);

  // 5) softmax with norms + per-batch temperature
  softmax_attn<<<dim3(2), 256, 0, stream>>>(partial, invn, temp, attn);

  // 6) fuse attn@v with output projection into W_eff[b]
  build_weff<<<dim3(B_), 256, 0, stream>>>(w_proj, attn, weff);

  // 7) out[b] = W_eff[b] (64x64) @ v[b] (64xS)           [WMMA f32]
  dim3 g7(S_ / (16 * NT * WPB), C_ / 16, B_);
  pw_gemm_wmma<<<g7, 256, 0, stream>>>(weff, 4096L, qkv + (long)128 * S_, (long)C3_ * S_,
                                       out, (long)C_ * S_);
}